// FB_Seg_90950227460831
// MI455X (gfx1250) — compile-verified
//
#include <hip/hip_runtime.h>
#include <math.h>

#define NLID  100000
#define NRAD  20000
#define NPTS  120000
#define MAPHW (512*512)
#define CHW   (64*MAPHW)
#define TILE  256
#define THREADS 512
#define TPB   469      // ceil(120000/256) tiles per batch

// activation row strides (floats): stride % 64 == 4 -> A-frag b64 loads hit 32 distinct banks
#define SX  132
#define SF  68
// pair-interleaved weight strides (floats per K-pair row): stride % 64 == 32 -> lane-half
// bank windows disjoint; each B fragment is one contiguous ds_load_b64
#define PSW 160        // 64-wide weights (wlin, w1): 64*2 data + 32 pad
#define PS2 96         // w2 (32-wide): 32*2 data + 32 pad
#define PS3 96         // w3 (16-wide): keep same modulus

// ---- single static LDS arena (static LDS starts at offset 0 -> async VDST offsets are literal) ----
#define OFF_X     0
#define OFF_FF    (OFF_X    + TILE * SX)        // 33792
#define OFF_WLIN  (OFF_FF   + TILE * SF)        // 51200
#define OFF_W1    (OFF_WLIN + 64 * PSW)         // 61440
#define OFF_W2    (OFF_W1   + 64 * PSW)         // 71680
#define OFF_W3    (OFF_W2   + 32 * PS2)         // 74752
#define OFF_W4    (OFF_W3   + 16 * PS3)         // 76288
#define OFF_BLIN  (OFF_W4   + 16)               // 76304
#define OFF_B1    (OFF_BLIN + 64)               // 76368
#define OFF_B2    (OFF_B1   + 64)               // 76432
#define OFF_B3    (OFF_B2   + 32)               // 76464
#define OFF_B4    (OFF_B3   + 16)               // 76480
#define OFF_COFF  (OFF_B4   + 4)                // 76484 (uint byte-offsets per point)
#define SMEM_FLOATS (OFF_COFF + TILE)           // 76740 floats = 306,960 B  (< 320 KB)

typedef float v2f __attribute__((ext_vector_type(2)));
typedef float v8f __attribute__((ext_vector_type(8)));

__device__ __forceinline__ v8f wmma4(v2f a, v2f b, v8f c) {
    // D = A(16x4,f32) * B(4x16,f32) + C(16x16,f32)
    return __builtin_amdgcn_wmma_f32_16x16x4_f32(
        false, a, false, b, (short)0, c, false, false);
}

__device__ __forceinline__ float gelu_exact(float x) {
    return 0.5f * x * (1.0f + erff(x * 0.70710678118654752f));
}

__global__ __launch_bounds__(THREADS, 1)
void fb_seg_fused(const float* __restrict__ pc0, const float* __restrict__ pc1,
                  const float* __restrict__ flow,
                  const long long* __restrict__ lidar,
                  const long long* __restrict__ radar,
                  const float* __restrict__ wlin, const float* __restrict__ blin,
                  const float* __restrict__ w1,   const float* __restrict__ b1,
                  const float* __restrict__ w2,   const float* __restrict__ b2,
                  const float* __restrict__ w3,   const float* __restrict__ b3,
                  const float* __restrict__ w4,   const float* __restrict__ b4,
                  float* __restrict__ out)
{
    __shared__ __align__(16) float smem[SMEM_FLOATS];

    float*    sX    = smem + OFF_X;     // [f0|f1] -> later pc (cols 0..63), h2 (cols 0..31)
    float*    sFF   = smem + OFF_FF;    // ff      -> later h1 (cols 0..63), h3 (cols 0..15)
    float*    sWlin = smem + OFF_WLIN;  // pair-interleaved
    float*    sW1   = smem + OFF_W1;
    float*    sW2   = smem + OFF_W2;
    float*    sW3   = smem + OFF_W3;
    float*    sW4   = smem + OFF_W4;
    float*    sBlin = smem + OFF_BLIN;
    float*    sB1   = smem + OFF_B1;
    float*    sB2   = smem + OFF_B2;
    float*    sB3   = smem + OFF_B3;
    float*    sB4   = smem + OFF_B4;
    unsigned* sOffB = (unsigned*)(smem + OFF_COFF);

    const int tid  = threadIdx.x;
    const int b    = blockIdx.x / TPB;
    const int tile = blockIdx.x % TPB;
    const int base = tile * TILE;

    // ---------------- phase 0: coords + packed weights into LDS ----------------
    if (tid < TILE) {
        int gp = base + tid;
        int r = 0, c = 0;
        if (gp < NPTS) {
            if (gp < NLID) {
                const long long* pco = lidar + ((long long)b * NLID + gp) * 3;
                r = (int)pco[1]; c = (int)pco[2];
            } else {
                const long long* pco = radar + ((long long)b * NRAD + (gp - NLID)) * 3;
                r = (int)pco[1]; c = (int)pco[2];
            }
        }
        sOffB[tid] = (unsigned)((r * 512 + c) * 4);   // spatial byte offset
    }
    for (int i = tid; i < 128 * 64; i += THREADS) {   // wlin, w1: (128,64) row-major in HBM
        int k = i >> 6, n = i & 63;
        int d = (k >> 1) * PSW + n * 2 + (k & 1);
        sWlin[d] = wlin[i];
        sW1  [d] = w1[i];
    }
    for (int i = tid; i < 64 * 32; i += THREADS) {    // w2: (64,32)
        int k = i >> 5, n = i & 31;
        sW2[(k >> 1) * PS2 + n * 2 + (k & 1)] = w2[i];
    }
    for (int i = tid; i < 32 * 16; i += THREADS) {    // w3: (32,16)
        int k = i >> 4, n = i & 15;
        sW3[(k >> 1) * PS3 + n * 2 + (k & 1)] = w3[i];
    }
    if (tid < 16) sW4[tid]   = w4[tid];
    if (tid < 64) sBlin[tid] = blin[tid];
    if (tid < 64) sB1[tid]   = b1[tid];
    if (tid < 32) sB2[tid]   = b2[tid];
    if (tid < 16) sB3[tid]   = b3[tid];
    if (tid == 0) sB4[0]     = b4[0];
    __syncthreads();

    // ---------------- phase 1: async gather straight into LDS (HBM-bound) ----------------
    {
        const float* p0b = pc0  + (size_t)b * CHW;
        const float* p1b = pc1  + (size_t)b * CHW;
        const float* flb = flow + (size_t)b * CHW;
        for (int i = tid; i < TILE * 64; i += THREADS) {
            int p  = i & (TILE - 1);
            int ch = i >> 8;                                    // wave-uniform
            unsigned goff = sOffB[p] + (unsigned)ch * (MAPHW * 4u);
            unsigned lX0  = (unsigned)((OFF_X + p * SX + ch) * 4);
            unsigned lX1  = lX0 + 64 * 4;
            unsigned lF   = (unsigned)((OFF_FF + p * SF + ch) * 4);
            asm volatile("global_load_async_to_lds_b32 %0, %1, %2"
                         :: "v"(lX0), "v"(goff), "s"(p0b) : "memory");
            asm volatile("global_load_async_to_lds_b32 %0, %1, %2"
                         :: "v"(lX1), "v"(goff), "s"(p1b) : "memory");
            asm volatile("global_load_async_to_lds_b32 %0, %1, %2"
                         :: "v"(lF),  "v"(goff), "s"(flb) : "memory");
        }
        asm volatile("s_wait_asynccnt 0x0" ::: "memory");
    }
    __syncthreads();

    // ---------------- phase 2: per-wave WMMA MLP on 16 rows ----------------
    const int lane = tid & 31;
    const int wv   = tid >> 5;
    const int p0   = wv * 16;
    const int colL = lane & 15;        // N index / A row index
    const int kh   = lane >> 4;        // lane-half K striping
    const int arow = p0 + colL;
    const int c2   = colL * 2;         // packed-weight column offset

    v8f acc[4];

    // ---- L-lin: pc = [f0|f1](128) @ wlin + blin  (no activation) ----
    #pragma unroll
    for (int nt = 0; nt < 4; ++nt) {
        float bv = sBlin[nt * 16 + colL];
        #pragma unroll
        for (int v = 0; v < 8; ++v) acc[nt][v] = bv;
    }
    for (int k0 = 0; k0 < 128; k0 += 4) {
        int kp = (k0 >> 1) + kh;
        v2f a = *(const v2f*)&sX[arow * SX + k0 + 2 * kh];
        #pragma unroll
        for (int nt = 0; nt < 4; ++nt) {
            v2f bb = *(const v2f*)&sWlin[kp * PSW + nt * 32 + c2];
            acc[nt] = wmma4(a, bb, acc[nt]);
        }
    }
    #pragma unroll
    for (int nt = 0; nt < 4; ++nt)
        #pragma unroll
        for (int v = 0; v < 8; ++v)
            sX[(p0 + v + 8 * kh) * SX + nt * 16 + colL] = acc[nt][v];   // pc -> X cols 0..63

    // ---- L1: h1 = gelu(pc @ w1[0:64] + ff @ w1[64:128] + b1) ----
    #pragma unroll
    for (int nt = 0; nt < 4; ++nt) {
        float bv = sB1[nt * 16 + colL];
        #pragma unroll
        for (int v = 0; v < 8; ++v) acc[nt][v] = bv;
    }
    for (int k0 = 0; k0 < 64; k0 += 4) {
        int kp = (k0 >> 1) + kh;
        v2f a  = *(const v2f*)&sX [arow * SX + k0 + 2 * kh];   // pc
        v2f a2 = *(const v2f*)&sFF[arow * SF + k0 + 2 * kh];   // ff
        #pragma unroll
        for (int nt = 0; nt < 4; ++nt) {
            v2f bb  = *(const v2f*)&sW1[kp * PSW + nt * 32 + c2];         // rows 0..63
            v2f bb2 = *(const v2f*)&sW1[(32 + kp) * PSW + nt * 32 + c2];  // rows 64..127
            acc[nt] = wmma4(a,  bb,  acc[nt]);
            acc[nt] = wmma4(a2, bb2, acc[nt]);
        }
    }
    #pragma unroll
    for (int nt = 0; nt < 4; ++nt)
        #pragma unroll
        for (int v = 0; v < 8; ++v)
            sFF[(p0 + v + 8 * kh) * SF + nt * 16 + colL] = gelu_exact(acc[nt][v]); // h1 -> FF

    // ---- L2: h2 = gelu(h1 @ w2 + b2)  (64 -> 32) ----
    #pragma unroll
    for (int nt = 0; nt < 2; ++nt) {
        float bv = sB2[nt * 16 + colL];
        #pragma unroll
        for (int v = 0; v < 8; ++v) acc[nt][v] = bv;
    }
    for (int k0 = 0; k0 < 64; k0 += 4) {
        int kp = (k0 >> 1) + kh;
        v2f a = *(const v2f*)&sFF[arow * SF + k0 + 2 * kh];
        #pragma unroll
        for (int nt = 0; nt < 2; ++nt) {
            v2f bb = *(const v2f*)&sW2[kp * PS2 + nt * 32 + c2];
            acc[nt] = wmma4(a, bb, acc[nt]);
        }
    }
    #pragma unroll
    for (int nt = 0; nt < 2; ++nt)
        #pragma unroll
        for (int v = 0; v < 8; ++v)
            sX[(p0 + v + 8 * kh) * SX + nt * 16 + colL] = gelu_exact(acc[nt][v]); // h2 -> X cols 0..31

    // ---- L3: h3 = gelu(h2 @ w3 + b3)  (32 -> 16) ----
    {
        float bv = sB3[colL];
        #pragma unroll
        for (int v = 0; v < 8; ++v) acc[0][v] = bv;
    }
    for (int k0 = 0; k0 < 32; k0 += 4) {
        int kp = (k0 >> 1) + kh;
        v2f a  = *(const v2f*)&sX[arow * SX + k0 + 2 * kh];
        v2f bb = *(const v2f*)&sW3[kp * PS3 + c2];
        acc[0] = wmma4(a, bb, acc[0]);
    }
    #pragma unroll
    for (int v = 0; v < 8; ++v)
        sFF[(p0 + v + 8 * kh) * SF + colL] = gelu_exact(acc[0][v]);     // h3 -> FF cols 0..15

    // ---- L4: s = sigmoid(h3 @ w4 + b4)  (16 -> 1), VALU epilogue ----
    if (lane < 16) {
        int p = p0 + lane;
        float s = sB4[0];
        #pragma unroll
        for (int j = 0; j < 16; ++j) s += sFF[p * SF + j] * sW4[j];
        s = 1.0f / (1.0f + expf(-s));
        int gp = base + p;
        if (gp < NPTS) out[(size_t)b * NPTS + gp] = s;
    }
}

extern "C" void kernel_launch(void* const* d_in, const int* in_sizes, int n_in,
                              void* d_out, int out_size, void* d_ws, size_t ws_size,
                              hipStream_t stream) {
    const float*     pc0   = (const float*)d_in[0];
    const float*     pc1   = (const float*)d_in[1];
    const float*     flow  = (const float*)d_in[2];
    const long long* lidar = (const long long*)d_in[3];
    const long long* radar = (const long long*)d_in[4];
    const float*     wlin  = (const float*)d_in[5];
    const float*     blin  = (const float*)d_in[6];
    const float*     w1    = (const float*)d_in[7];
    const float*     b1    = (const float*)d_in[8];
    const float*     w2    = (const float*)d_in[9];
    const float*     b2    = (const float*)d_in[10];
    const float*     w3    = (const float*)d_in[11];
    const float*     b3    = (const float*)d_in[12];
    const float*     w4    = (const float*)d_in[13];
    const float*     b4    = (const float*)d_in[14];
    float*           out   = (float*)d_out;

    dim3 grid(2 * TPB);        // 2 batches x 469 tiles of 256 points
    dim3 block(THREADS);
    fb_seg_fused<<<grid, block, 0, stream>>>(pc0, pc1, flow, lidar, radar,
                                             wlin, blin, w1, b1, w2, b2,
                                             w3, b3, w4, b4, out);
}